// GCNII_model_19318762897563
// MI455X (gfx1250) — compile-verified
//
#include <hip/hip_runtime.h>
#include <math.h>

// GCNII forward on MI455X (gfx1250, wave32).
// fp32 WMMA path: V_WMMA_F32_16X16X4_F32 (keeps reference fp32 precision).
// Input-GEMM staging uses double-buffered GLOBAL_LOAD_ASYNC_TO_LDS (ASYNCcnt):
// chunk i+1 streams into the alternate LDS buffer while chunk i is consumed by
// WMMA; s_wait_asynccnt 6 exploits in-order ASYNCcnt decrement (6 async
// instructions per wave per chunk) to wait only for the older chunk.

typedef float v2f __attribute__((ext_vector_type(2)));
typedef float v8f __attribute__((ext_vector_type(8)));
typedef int   b128_t __attribute__((vector_size(16)));   // matches builtin param type

#define HD 64      // hidden dim (reference hardcodes H=64)
#define KC 32      // K-chunk for the input GEMM
#define SA_STR 36  // LDS stride for A tile rows (conflict-free spread)
#define SB_STR 68  // LDS stride for B chunk rows (16B-aligned rows for b128)
#define SL_STR 68  // LDS stride for 64-wide layer tiles (4r mod 64 distinct)

__device__ __forceinline__ v8f wmma_f32x4(v2f a, v2f b, v8f c) {
  // 8 args: (neg_a, A, neg_b, B, c_mod, C, reuse_a, reuse_b)
  return __builtin_amdgcn_wmma_f32_16x16x4_f32(
      false, a, false, b, (short)0, c, false, false);
}

// Async global -> LDS copies (gfx1250, tracked by ASYNCcnt).
__device__ __forceinline__ void async_cp_b32(const float* g, float* l) {
  __builtin_amdgcn_global_load_async_to_lds_b32(
      (__attribute__((address_space(1))) int*)g,
      (__attribute__((address_space(3))) int*)l, 0, 0);
}
__device__ __forceinline__ void async_cp_b128(const float* g, float* l) {
  __builtin_amdgcn_global_load_async_to_lds_b128(
      (__attribute__((address_space(1))) b128_t*)g,
      (__attribute__((address_space(3))) b128_t*)l, 0, 0);
}

// ---------------- elementwise / graph kernels ----------------

__global__ __launch_bounds__(256) void k_zero(float* __restrict__ p, int n4) {
  int i = blockIdx.x * blockDim.x + threadIdx.x;
  if (i < n4) ((float4*)p)[i] = make_float4(0.f, 0.f, 0.f, 0.f);
}

__global__ __launch_bounds__(256) void k_deg(const int* __restrict__ dst,
                                             float* __restrict__ deg, int E) {
  int e = blockIdx.x * blockDim.x + threadIdx.x;
  if (e < E) atomicAdd(&deg[dst[e]], 1.f);
}

__global__ __launch_bounds__(256) void k_rsqrt(float* __restrict__ deg, int n) {
  int i = blockIdx.x * blockDim.x + threadIdx.x;
  if (i < n) {
    float d = deg[i];
    deg[i] = d > 0.f ? rsqrtf(d) : 0.f;
  }
}

__global__ __launch_bounds__(256) void k_norm(const int* __restrict__ src,
                                              const int* __restrict__ dst,
                                              const float* __restrict__ dinv,
                                              float* __restrict__ norm, int E) {
  int e = blockIdx.x * blockDim.x + threadIdx.x;
  if (e < E) norm[e] = dinv[src[e]] * dinv[dst[e]];
}

// One wave per edge; each lane handles features lane and lane+32.
// h (12.8 MB) is L2-resident -> gathers hit L2; scatters are L2 f32 atomics.
__global__ __launch_bounds__(256) void k_spmm(const int* __restrict__ src,
                                              const int* __restrict__ dst,
                                              const float* __restrict__ norm,
                                              const float* __restrict__ h,
                                              float* __restrict__ agg, int E) {
  int gid = blockIdx.x * blockDim.x + threadIdx.x;
  int e = gid >> 5;
  int lane = threadIdx.x & 31;
  if (e >= E) return;
  int s = src[e], d = dst[e];
  float w = norm[e];
  const float* hs = h + (size_t)s * HD;
  float* ad = agg + (size_t)d * HD;
  atomicAdd(&ad[lane],      w * hs[lane]);
  atomicAdd(&ad[lane + 32], w * hs[lane + 32]);
}

// ---------------- input GEMM: h = relu(x @ W0 + b0), also h0 = h ----------------
// Block = 8 waves = 2 M-tiles x 4 N-tiles (32x64 output). Double-buffered async
// staging: while chunk i is consumed by 8 WMMAs/wave, chunk i+1 streams into
// the other LDS buffer. Addresses are clamped (branch-free); B rows past K are
// zeroed so clamped-A garbage contributes nothing.
__global__ __launch_bounds__(256) void k_gemm0(const float* __restrict__ x,
                                               const float* __restrict__ W0,
                                               const float* __restrict__ b0,
                                               float* __restrict__ h,
                                               float* __restrict__ h0,
                                               int N, int K) {
  __shared__ float sA[2][32 * SA_STR];   // A tiles: [row][k]
  __shared__ float sB[2][KC * SB_STR];   // W0 chunks: [k][col], padded rows
  int tid  = threadIdx.x;
  int wave = tid >> 5;
  int lane = tid & 31;
  int r  = lane & 15;
  int kh = lane >> 4;
  int mrow0 = blockIdx.x * 32;
  int lm = (wave >> 2) * 16;             // local M base (0 or 16)
  int n0 = (wave & 3) * 16;              // N tile base

  // Issue one chunk's async copies: 4x b32 (A) + 2x b128 (B) per thread.
  auto issue_chunk = [&](int k0, float* dA, float* dB) {
    #pragma unroll
    for (int q = 0; q < 4; ++q) {
      int i  = tid + q * 256;            // [0, 1024)
      int rr = i >> 5;                   // 0..31
      int cc = i & 31;                   // 0..31
      int gr = mrow0 + rr; if (gr > N - 1) gr = N - 1;
      int kk = k0 + cc;    if (kk > K - 1) kk = K - 1;
      async_cp_b32(x + (size_t)gr * K + kk, &dA[rr * SA_STR + cc]);
    }
    #pragma unroll
    for (int q = 0; q < 2; ++q) {
      int sg = tid + q * 256;            // [0, 512)
      int rr = sg >> 4;                  // 0..31 (k within chunk)
      int sc = (sg & 15) * 4;            // col base 0,4,...,60
      int kk = k0 + rr; if (kk > K - 1) kk = K - 1;
      async_cp_b128(W0 + (size_t)kk * HD + sc, &dB[rr * SB_STR + sc]);
    }
  };

  int nch = (K + KC - 1) / KC;
  v8f acc = {};
  issue_chunk(0, sA[0], sB[0]);
  for (int ch = 0; ch < nch; ++ch) {
    int cur = ch & 1;
    if (ch + 1 < nch) {
      issue_chunk((ch + 1) * KC, sA[cur ^ 1], sB[cur ^ 1]);
      // ASYNCcnt decrements in issue order; <=6 outstanding => chunk ch landed.
      __builtin_amdgcn_s_wait_asynccnt(6);
    } else {
      __builtin_amdgcn_s_wait_asynccnt(0);
    }
    __syncthreads();
    int k0 = ch * KC;
    if (k0 + KC > K) {                   // K-tail: zero B rows past K (uniform)
      int vrows = K - k0;
      for (int i = tid; i < (KC - vrows) * HD; i += 256)
        sB[cur][(vrows + (i >> 6)) * SB_STR + (i & 63)] = 0.f;
      __syncthreads();
    }
    const float* cA = sA[cur];
    const float* cB = sB[cur];
    #pragma unroll
    for (int ks = 0; ks < KC; ks += 4) {
      int ka = ks + 2 * kh;
      v2f a = *(const v2f*)&cA[(lm + r) * SA_STR + ka];
      v2f b;
      b.x = cB[ka       * SB_STR + n0 + r];
      b.y = cB[(ka + 1) * SB_STR + n0 + r];
      acc = wmma_f32x4(a, b, acc);
    }
    __syncthreads();                     // buffer cur free for reissue
  }

  float bc = b0[n0 + r];
  int obase = mrow0 + lm;                // C/D layout: VGPR j -> M=j (lo) / j+8 (hi)
  if (mrow0 + 32 <= N) {                 // uniform: full block, unguarded epilogue
    #pragma unroll
    for (int j = 0; j < 8; ++j) {
      float v = acc[j] + bc;
      v = v > 0.f ? v : 0.f;
      size_t idx = (size_t)(obase + j + 8 * kh) * HD + n0 + r;
      h[idx]  = v;
      h0[idx] = v;
    }
  } else {
    #pragma unroll
    for (int j = 0; j < 8; ++j) {
      int orow = obase + j + 8 * kh;
      if (orow < N) {
        float v = acc[j] + bc;
        v = v > 0.f ? v : 0.f;
        size_t idx = (size_t)orow * HD + n0 + r;
        h[idx]  = v;
        h0[idx] = v;
      }
    }
  }
}

// ---------------- layer: h = relu(beta*(S@W) + (1-beta)*S), S = 0.9*agg + 0.1*h0 ----------------
// Tiles are computed/transposed, so staged through VGPRs; 16 unrolled WMMAs per
// wave, all operands via ds_load_b64.
__global__ __launch_bounds__(256) void k_layer(const float* __restrict__ agg,
                                               const float* __restrict__ h0,
                                               const float* __restrict__ W,
                                               float beta,
                                               float* __restrict__ h, int N) {
  __shared__ float sWt[HD * SL_STR];   // W transposed: [col][k]
  __shared__ float sS [32 * SL_STR];   // fused support tile: [row][k]
  int tid  = threadIdx.x;
  int mblk = blockIdx.x * 32;

  // stage W transposed (coalesced reads; i over cols fastest)
  #pragma unroll
  for (int q = 0; q < 16; ++q) {
    int i = tid + q * 256;             // [0, 4096)
    int k = i >> 6, col = i & 63;
    sWt[col * SL_STR + k] = W[i];
  }
  // stage fused support tile with clamped rows (branch-free)
  #pragma unroll
  for (int q = 0; q < 8; ++q) {
    int i  = tid + q * 256;            // [0, 2048)
    int rr = i >> 6, cc = i & 63;
    int grow = mblk + rr;
    int gc   = grow < N ? grow : N - 1;
    size_t idx = (size_t)gc * HD + cc;
    sS[rr * SL_STR + cc] = 0.9f * agg[idx] + 0.1f * h0[idx]; // (1-ALPHA), ALPHA
  }
  __syncthreads();

  int wave = tid >> 5, lane = tid & 31;
  int lm = (wave >> 2) * 16;
  int n0 = (wave & 3) * 16;
  int r  = lane & 15, kh = lane >> 4;

  v8f acc = {};
  #pragma unroll
  for (int ks = 0; ks < HD; ks += 4) {
    int ka = ks + 2 * kh;
    v2f a = *(const v2f*)&sS [(lm + r) * SL_STR + ka];
    v2f b = *(const v2f*)&sWt[(n0 + r) * SL_STR + ka];
    acc = wmma_f32x4(a, b, acc);
  }
  float ob = 1.f - beta;
  if (mblk + 32 <= N) {                // uniform: unguarded epilogue
    #pragma unroll
    for (int j = 0; j < 8; ++j) {
      int lrow = lm + j + 8 * kh;
      float s = sS[lrow * SL_STR + n0 + r];
      float v = beta * acc[j] + ob * s;
      h[(size_t)(mblk + lrow) * HD + n0 + r] = v > 0.f ? v : 0.f;
    }
  } else {
    #pragma unroll
    for (int j = 0; j < 8; ++j) {
      int lrow = lm + j + 8 * kh;
      int orow = mblk + lrow;
      if (orow < N) {
        float s = sS[lrow * SL_STR + n0 + r];
        float v = beta * acc[j] + ob * s;
        h[(size_t)orow * HD + n0 + r] = v > 0.f ? v : 0.f;
      }
    }
  }
}

// ---------------- head: log_softmax(h @ W_out + b_out) ----------------
#define MAXC 16
__global__ __launch_bounds__(256) void k_head(const float* __restrict__ h,
                                              const float* __restrict__ Wo,
                                              const float* __restrict__ bo,
                                              float* __restrict__ out,
                                              int N, int C) {
  int i = blockIdx.x * blockDim.x + threadIdx.x;
  if (i >= N) return;
  const float* hi = h + (size_t)i * HD;
  float logit[MAXC];
  float m = -1e30f;
  for (int c = 0; c < C; ++c) {
    float acc = bo[c];
    #pragma unroll
    for (int k = 0; k < HD; ++k) acc += hi[k] * Wo[k * C + c];
    logit[c] = acc;
    m = acc > m ? acc : m;
  }
  float sum = 0.f;
  for (int c = 0; c < C; ++c) sum += __expf(logit[c] - m);
  float lse = m + __logf(sum);
  float* oi = out + (size_t)i * C;
  for (int c = 0; c < C; ++c) oi[c] = logit[c] - lse;
}

// ---------------- host ----------------

extern "C" void kernel_launch(void* const* d_in, const int* in_sizes, int n_in,
                              void* d_out, int out_size, void* d_ws, size_t ws_size,
                              hipStream_t stream) {
  const float* x    = (const float*)d_in[0];
  const int*   eidx = (const int*)  d_in[1];
  const float* W0   = (const float*)d_in[2];
  const float* b0   = (const float*)d_in[3];
  const float* Ws   = (const float*)d_in[4];
  const float* Wout = (const float*)d_in[5];
  const float* bout = (const float*)d_in[6];

  const int Hh = in_sizes[3];             // 64
  const int F  = in_sizes[2] / Hh;        // 1433
  const int N  = in_sizes[0] / F;         // 50000
  const int E  = in_sizes[1] / 2;         // 1,650,000
  const int L  = in_sizes[4] / (Hh * Hh); // 8
  const int C  = in_sizes[5] / Hh;        // 7
  const int* src = eidx;
  const int* dst = eidx + E;

  // carve workspace (256B aligned slabs): dinv, norm, h, h0, agg  (~45 MB)
  char* ws = (char*)d_ws;
  size_t off = 0;
  auto carve = [&](size_t elems) {
    float* p = (float*)(ws + off);
    off += (elems * sizeof(float) + 255) & ~(size_t)255;
    return p;
  };
  float* dinv = carve((size_t)N);
  float* norm = carve((size_t)E);
  float* h    = carve((size_t)N * HD);
  float* h0   = carve((size_t)N * HD);
  float* agg  = carve((size_t)N * HD);
  (void)ws_size; (void)n_in; (void)out_size;

  // degrees -> dinv -> edge norms
  int n4 = (N + 3) / 4;
  hipLaunchKernelGGL(k_zero,  dim3((n4 + 255) / 256), dim3(256), 0, stream, dinv, n4);
  hipLaunchKernelGGL(k_deg,   dim3((E + 255) / 256),  dim3(256), 0, stream, dst, dinv, E);
  hipLaunchKernelGGL(k_rsqrt, dim3((N + 255) / 256),  dim3(256), 0, stream, dinv, N);
  hipLaunchKernelGGL(k_norm,  dim3((E + 255) / 256),  dim3(256), 0, stream, src, dst, dinv, norm, E);

  // input GEMM + bias + relu -> h, h0   (one block per 32 output rows)
  int nMT = (N + 15) / 16;
  hipLaunchKernelGGL(k_gemm0, dim3((nMT + 1) / 2), dim3(256), 0, stream,
                     x, W0, b0, h, h0, N, F);

  // 8 GCNII layers
  int aggN4 = (N * HD) / 4;
  long long spmmThreads = (long long)E * 32;
  for (int i = 0; i < L; ++i) {
    float beta = logf(0.5f / (float)(i + 1) + 1.f);   // log(LAMBDA/(i+1)+1)
    hipLaunchKernelGGL(k_zero, dim3((aggN4 + 255) / 256), dim3(256), 0, stream, agg, aggN4);
    hipLaunchKernelGGL(k_spmm, dim3((unsigned)((spmmThreads + 255) / 256)), dim3(256), 0, stream,
                       src, dst, norm, h, agg, E);
    hipLaunchKernelGGL(k_layer, dim3((nMT + 1) / 2), dim3(256), 0, stream,
                       agg, h0, Ws + (size_t)i * Hh * Hh, beta, h, N);
  }

  // head: logits + log_softmax
  hipLaunchKernelGGL(k_head, dim3((N + 255) / 256), dim3(256), 0, stream,
                     h, Wout, bout, (float*)d_out, N, C);
}